// VectorQuantizer_58222576664787
// MI455X (gfx1250) — compile-verified
//
#include <hip/hip_runtime.h>
#include <hip/hip_bf16.h>

typedef __attribute__((ext_vector_type(2))) float v2f;
typedef __attribute__((ext_vector_type(8))) float v8f;
typedef __attribute__((ext_vector_type(4))) int   v4i;

#define N_Q     16384   // 16*32*32 queries
#define VOCAB   16384
#define CCH     32
#define N_PIX   524288  // 16*32*32*32 elements of f / fhat

#define STAGE_CODES 128                       // codes staged in LDS per step
#define LDS_STRIDE  36                        // padded floats/row: 16B aligned, bank-conflict-free
#define NSTAGES     (VOCAB / STAGE_CODES)     // 128

#if __has_builtin(__builtin_amdgcn_global_load_async_to_lds_b128)
#define VQ_ASYNC 1
#else
#define VQ_ASYNC 0
#endif

static __device__ __forceinline__ void vq_wait_async() {
#if __has_builtin(__builtin_amdgcn_s_wait_asynccnt)
  __builtin_amdgcn_s_wait_asynccnt(0);
#else
  asm volatile("s_wait_asynccnt 0x0" ::: "memory");
#endif
}

// -------------------- workspace layout (bytes) --------------------
// [0)        idx     int[16384]
// [65536)    counts  int[16384]
// [131072)   enorm   float[16384]
// [196608)   zq      float[524288]   (BCHW)
// [2293760)  used    int
// [2293768)  bsum    float[2048]     per-block SSE partials
// ------------------------------------------------------------------

__global__ __launch_bounds__(256) void vq_enorm_kernel(
    const float* __restrict__ emb, float* __restrict__ enorm) {
  int v = blockIdx.x * 256 + threadIdx.x;
  const float4* p = (const float4*)(emb + v * CCH);
  float s = 0.f;
#pragma unroll
  for (int j = 0; j < 8; ++j) {
    float4 q = p[j];
    s += q.x * q.x + q.y * q.y + q.z * q.z + q.w * q.w;
  }
  enorm[v] = s;
}

__global__ __launch_bounds__(256) void vq_zero_kernel(
    int* __restrict__ counts, int* __restrict__ used) {
  int tid = blockIdx.x * 256 + threadIdx.x;
  if (tid < VOCAB) counts[tid] = 0;
  if (tid == 0) *used = 0;
}

// Fused distance-GEMM + argmin with LDS-staged codebook.
// Block = 8 waves, each wave owns 16 queries; the 8 waves share each
// 128-code LDS stage (8x reduction in global/L2 codebook traffic).
// dist = ||e||^2 - 2 q.e  (||q||^2 dropped: argmin-invariant per row)
__global__ __launch_bounds__(256) void vq_argmin_kernel(
    const float* __restrict__ f, const float* __restrict__ emb,
    const float* __restrict__ enorm, int* __restrict__ idx_out) {
  __shared__ float lbuf[2][STAGE_CODES * LDS_STRIDE];

  const int t     = threadIdx.x;
  const int lane  = t & 31;
  const int qtile = blockIdx.x * 8 + (t >> 5);  // 0..1023
  const int half  = lane >> 4;   // 0 -> K pair {0,1}, 1 -> {2,3} (A & B layout)
  const int l15   = lane & 15;

  // ---- A tile: 16 queries x 32 channels resident in 8 x v2f ----
  // query[n][c] = f[b, c, y, x], n = b*1024 + y*32 + x (channel stride 1024)
  const int row = qtile * 16 + l15;
  const int bb  = row >> 10;
  const int rem = row & 1023;
  const float* fq = f + bb * 32768 + rem;
  v2f a[8];
#pragma unroll
  for (int j = 0; j < 8; ++j) {
    const int kb = 4 * j + 2 * half;
    a[j].x = fq[kb * 1024];
    a[j].y = fq[(kb + 1) * 1024];
  }

  float minv[8];
  int   mini[8];
#pragma unroll
  for (int r = 0; r < 8; ++r) { minv[r] = 3.4e38f; mini[r] = 0; }

#if VQ_ASYNC
  // ---- ASYNCcnt path: global -> LDS DMA, b128 granularity ----
  auto stage_copy = [&](int s, int b) {
#pragma unroll
    for (int i = 0; i < 4; ++i) {
      int q   = i * 256 + t;          // 1024 float4s per stage
      int r0  = q >> 3;
      int c4  = (q & 7) * 4;
      __builtin_amdgcn_global_load_async_to_lds_b128(
          (v4i*)(emb + (s * STAGE_CODES + r0) * CCH + c4),
          (v4i*)(&lbuf[b][r0 * LDS_STRIDE + c4]),
          0, 0);
    }
  };
  stage_copy(0, 0);
  vq_wait_async();
  __syncthreads();
#else
  // ---- fallback: software-pipelined global->VGPR->ds_store copy ----
  v2f creg[8];
  auto stage_load = [&](int s) {
#pragma unroll
    for (int i = 0; i < 8; ++i) {
      int q  = i * 256 + t;           // 2048 v2fs per stage
      int r0 = q >> 4;
      int c2 = (q & 15) * 2;
      creg[i] = *(const v2f*)(emb + (s * STAGE_CODES + r0) * CCH + c2);
    }
  };
  auto stage_store = [&](int b) {
#pragma unroll
    for (int i = 0; i < 8; ++i) {
      int q  = i * 256 + t;
      int r0 = q >> 4;
      int c2 = (q & 15) * 2;
      *(v2f*)(&lbuf[b][r0 * LDS_STRIDE + c2]) = creg[i];
    }
  };
  stage_load(0);
  stage_store(0);
  __syncthreads();
#endif

  for (int s = 0; s < NSTAGES; ++s) {
    const float* lb = lbuf[s & 1];

#if VQ_ASYNC
    if (s + 1 < NSTAGES) stage_copy(s + 1, (s + 1) & 1);
#else
    if (s + 1 < NSTAGES) stage_load(s + 1);   // global loads in flight over compute
#endif

#pragma unroll
    for (int st = 0; st < 8; ++st) {          // 8 sub-tiles of 16 codes
      const int code = st * 16 + l15;
      const int col  = s * STAGE_CODES + code;
      const float* lp = lb + code * LDS_STRIDE + 2 * half;
      v8f c = {0.f, 0.f, 0.f, 0.f, 0.f, 0.f, 0.f, 0.f};
#pragma unroll
      for (int j = 0; j < 8; ++j) {           // K = 32 in 8 steps of 4
        v2f bm = *(const v2f*)(lp + 4 * j);
        c = __builtin_amdgcn_wmma_f32_16x16x4_f32(false, a[j], false, bm,
                                                  (short)0, c, false, false);
      }
      const float en = enorm[col];
#pragma unroll
      for (int r = 0; r < 8; ++r) {           // rows: half*8 + r
        const float d = en - 2.0f * c[r];
        if (d < minv[r]) { minv[r] = d; mini[r] = col; }
      }
    }

#if VQ_ASYNC
    vq_wait_async();                          // next-buffer DMA landed
#else
    if (s + 1 < NSTAGES) stage_store((s + 1) & 1);
#endif
    __syncthreads();
  }

  // cross-lane argmin within each 16-lane half (xor masks stay in-half)
#pragma unroll
  for (int r = 0; r < 8; ++r) {
    float v = minv[r];
    int   i = mini[r];
#pragma unroll
    for (int off = 8; off >= 1; off >>= 1) {
      float ov = __shfl_xor(v, off, 32);
      int   oi = __shfl_xor(i, off, 32);
      if (ov < v || (ov == v && oi < i)) { v = ov; i = oi; }
    }
    mini[r] = i;
  }
  if (l15 == 0) {
#pragma unroll
    for (int r = 0; r < 8; ++r)
      idx_out[qtile * 16 + half * 8 + r] = mini[r];
  }
}

// bincount + gather codebook rows into zq (BCHW)
__global__ __launch_bounds__(256) void vq_scatter_kernel(
    const int* __restrict__ idx, const float* __restrict__ emb,
    float* __restrict__ zq, int* __restrict__ counts) {
  int n = blockIdx.x * 256 + threadIdx.x;  // 0..16383
  int id = idx[n];
  atomicAdd(&counts[id], 1);
  int bb = n >> 10, rem = n & 1023;
  float* zb = zq + bb * 32768 + rem;
  const float* er = emb + id * CCH;
#pragma unroll
  for (int c = 0; c < CCH; ++c) zb[c * 1024] = er[c];
}

// fhat = 0.5*zq + 0.5*(conv3x3(zq)+bias); write fhat; per-block SSE partial
__global__ __launch_bounds__(256) void vq_conv_loss_kernel(
    const float* __restrict__ zq, const float* __restrict__ f,
    const float* __restrict__ w, const float* __restrict__ bias,
    float* __restrict__ out, float* __restrict__ bsum) {
  int tid = blockIdx.x * 256 + threadIdx.x;     // BCHW flat
  int x  = tid & 31;
  int y  = (tid >> 5) & 31;
  int oc = (tid >> 10) & 31;
  int bb = tid >> 15;
  const float* zb = zq + bb * 32768;
  const float* wp = w + oc * (CCH * 9);
  float acc = bias[oc];
  for (int ic = 0; ic < CCH; ++ic) {
    const float* zc = zb + ic * 1024;
    const float* wc = wp + ic * 9;
#pragma unroll
    for (int ky = 0; ky < 3; ++ky) {
      int yy = y + ky - 1;
      if ((unsigned)yy < 32u) {
#pragma unroll
        for (int kx = 0; kx < 3; ++kx) {
          int xx = x + kx - 1;
          if ((unsigned)xx < 32u) acc += wc[ky * 3 + kx] * zc[yy * 32 + xx];
        }
      }
    }
  }
  float fh = 0.5f * (zb[oc * 1024 + y * 32 + x] + acc);
  out[tid] = fh;                                 // straight-through fwd value
  float d = fh - f[tid];

  __shared__ float red[256];
  red[threadIdx.x] = d * d;
  __syncthreads();
  for (int s = 128; s > 0; s >>= 1) {
    if (threadIdx.x < s) red[threadIdx.x] += red[threadIdx.x + s];
    __syncthreads();
  }
  if (threadIdx.x == 0) bsum[blockIdx.x] = red[0];
}

__global__ __launch_bounds__(256) void vq_usage_kernel(
    const int* __restrict__ counts, int* __restrict__ used) {
  int v = blockIdx.x * 256 + threadIdx.x;
  // prob > 0.01/V  <=>  count > 0.01  <=>  count >= 1
  if (counts[v] > 0) atomicAdd(used, 1);
}

__global__ void vq_final_kernel(const float* __restrict__ bsum,
                                const int* __restrict__ used,
                                float* __restrict__ out) {
  if (threadIdx.x == 0 && blockIdx.x == 0) {
    float s = 0.f;
    for (int i = 0; i < 2048; ++i) s += bsum[i];   // deterministic fixed order
    out[N_PIX]     = 1.25f * s / (float)N_PIX;     // (BETA + 1) * mean
    out[N_PIX + 1] = (float)(*used) * (100.0f / (float)VOCAB);
  }
}

extern "C" void kernel_launch(void* const* d_in, const int* in_sizes, int n_in,
                              void* d_out, int out_size, void* d_ws, size_t ws_size,
                              hipStream_t stream) {
  const float* f      = (const float*)d_in[0];   // [16,32,32,32]
  const float* emb    = (const float*)d_in[1];   // [16384,32]
  const float* conv_w = (const float*)d_in[2];   // [32,32,3,3]
  const float* conv_b = (const float*)d_in[3];   // [32]
  float* out = (float*)d_out;

  char* ws = (char*)d_ws;
  int*   idx    = (int*)(ws);
  int*   counts = (int*)(ws + 65536);
  float* enorm  = (float*)(ws + 131072);
  float* zq     = (float*)(ws + 196608);
  int*   used   = (int*)(ws + 2293760);
  float* bsum   = (float*)(ws + 2293768);

  vq_enorm_kernel<<<VOCAB / 256, 256, 0, stream>>>(emb, enorm);
  vq_zero_kernel<<<VOCAB / 256, 256, 0, stream>>>(counts, used);
  vq_argmin_kernel<<<128, 256, 0, stream>>>(f, emb, enorm, idx);
  vq_scatter_kernel<<<N_Q / 256, 256, 0, stream>>>(idx, emb, zq, counts);
  vq_conv_loss_kernel<<<N_PIX / 256, 256, 0, stream>>>(zq, f, conv_w, conv_b, out, bsum);
  vq_usage_kernel<<<VOCAB / 256, 256, 0, stream>>>(counts, used);
  vq_final_kernel<<<1, 32, 0, stream>>>(bsum, used, out);
}